// UnifiedMultiLevel_12137577578975
// MI455X (gfx1250) — compile-verified
//
#include <hip/hip_runtime.h>
#include <hip/hip_bf16.h>

typedef float v2f __attribute__((ext_vector_type(2)));
typedef float v8f __attribute__((ext_vector_type(8)));

#define TPB    256
#define MROWS  32
#define NCOLS  128
#define KC     16      // Chebyshev input nodes per K-chunk (K-per-chunk = 64)
#define KPAD   4

// out[r][o] = Add[r][o] + sign * gelu( sum_{i,d} T_d(tanh(X[r][i])) * Cf[i][o][d] )
// optional: GOut[r][o] = out[r][o] * sigmoid(impw[wIdx])
__global__ __launch_bounds__(TPB) void kan_lift_kernel(
    const float* __restrict__ X, long xR, long xC,
    const float* __restrict__ Cf,
    const float* __restrict__ Add, long aR, long aC,
    float sign,
    float* __restrict__ Out, long oR, long oC,
    float* __restrict__ GOut,
    const float* __restrict__ impw, int wIdx,
    int DM)
{
    __shared__ float sA[MROWS][KC * 4 + KPAD];   // basis, row-major per M-row
    __shared__ float sB[NCOLS][KC * 4 + KPAD];   // coeffs transposed: [n][k]

    const int tid   = threadIdx.x;
    const int wave  = tid >> 5;
    const int lane  = tid & 31;
    const int mtile = wave >> 2;          // 0..1  (16-row tile)
    const int npair = wave & 3;           // 0..3  (32-col slab; two 16-col tiles)
    const long mBase = (long)blockIdx.y * MROWS;
    const long nBase = (long)blockIdx.x * NCOLS;

    v8f acc0 = {0.f, 0.f, 0.f, 0.f, 0.f, 0.f, 0.f, 0.f};
    v8f acc1 = {0.f, 0.f, 0.f, 0.f, 0.f, 0.f, 0.f, 0.f};

    const int kp    = (lane >> 4) << 1;             // 0 or 2 (K-pair select)
    const int aRow  = mtile * 16 + (lane & 15);     // A fragment source row
    const int bRow0 = npair * 32 + (lane & 15);     // B fragment col (n tile 0)
    const int bRow1 = bRow0 + 16;                   // B fragment col (n tile 1)

    for (int i0 = 0; i0 < DM; i0 += KC) {
        // ---- stage Chebyshev basis (32 rows x 16 nodes) ----
        for (int t = tid; t < MROWS * KC; t += TPB) {
            int r  = t >> 4;           // KC == 16
            int ii = t & 15;
            float xv = X[(mBase + r) * xR + (long)(i0 + ii) * xC];
            float tt = tanhf(xv);
            tt = fminf(1.f, fmaxf(-1.f, tt));
            float t2 = 2.f * tt * tt - 1.f;          // T2
            float t3 = tt * (2.f * t2 - 1.f);        // T3 = 4t^3-3t
            sA[r][ii * 4 + 0] = 1.f;
            sA[r][ii * 4 + 1] = tt;
            sA[r][ii * 4 + 2] = t2;
            sA[r][ii * 4 + 3] = t3;
        }
        // ---- stage coeffs: per node ii, 128 outputs x 4 degrees contiguous ----
        #pragma unroll
        for (int ii = 0; ii < KC; ++ii) {
            const long rowBase = (long)(i0 + ii) * DM * 4 + nBase * 4;
            #pragma unroll
            for (int e = tid; e < NCOLS * 4; e += TPB) {
                sB[e >> 2][ii * 4 + (e & 3)] = Cf[rowBase + e];
            }
        }
        if (i0 + KC < DM) {  // gfx1250 global_prefetch_b8 on next coeff slab
            __builtin_prefetch(&Cf[(long)(i0 + KC) * DM * 4 + nBase * 4 + tid * 8], 0, 1);
        }
        __syncthreads();

        #pragma unroll
        for (int kk = 0; kk < KC * 4; kk += 4) {
            v2f a  = *(const v2f*)&sA[aRow][kk + kp];
            v2f b0 = *(const v2f*)&sB[bRow0][kk + kp];
            v2f b1 = *(const v2f*)&sB[bRow1][kk + kp];
            acc0 = __builtin_amdgcn_wmma_f32_16x16x4_f32(
                       false, a, false, b0, (short)0, acc0, false, false);
            acc1 = __builtin_amdgcn_wmma_f32_16x16x4_f32(
                       false, a, false, b1, (short)0, acc1, false, false);
        }
        __syncthreads();
    }

    float w = 1.f;
    if (GOut != nullptr) w = 1.f / (1.f + expf(-impw[wIdx]));
    const int nl  = lane & 15;
    const int mhi = (lane >> 4) * 8;
    #pragma unroll
    for (int r = 0; r < 8; ++r) {
        long row  = mBase + mtile * 16 + mhi + r;
        long col0 = nBase + npair * 32 + nl;
        long col1 = col0 + 16;

        float g0v = acc0[r];
        float ge0 = 0.5f * g0v * (1.f + erff(g0v * 0.70710678118654752f)); // exact GELU
        float v0  = Add[row * aR + col0 * aC] + sign * ge0;
        Out[row * oR + col0 * oC] = v0;

        float g1v = acc1[r];
        float ge1 = 0.5f * g1v * (1.f + erff(g1v * 0.70710678118654752f));
        float v1  = Add[row * aR + col1 * aC] + sign * ge1;
        Out[row * oR + col1 * oC] = v1;

        if (GOut != nullptr) {
            GOut[row * (long)DM + col0] = v0 * w;
            GOut[row * (long)DM + col1] = v1 * w;
        }
    }
}

// per-batch |c|,|d| sums + global smoothL1(d), ||c||^2, ||xe||^2
__global__ __launch_bounds__(256) void reduce_level_kernel(
    const float* __restrict__ d, const float* __restrict__ c,
    const float* __restrict__ xe, long xeR, long xeC,
    int dm, float* __restrict__ acc)
{
    const int CL = 128 * dm;
    const int b   = blockIdx.x;
    const int tid = threadIdx.x;
    float sAbsC = 0.f, sAbsD = 0.f, sSl1 = 0.f, sC2 = 0.f, sXe2 = 0.f;
    const long base = (long)b * CL;
    for (int e = tid; e < CL; e += 256) {
        float dv = d[base + e];
        float cv = c[base + e];
        long  row = (long)b * 128 + e / dm;
        int   col = e % dm;
        float xv = xe[row * xeR + (long)col * xeC];
        float ad = fabsf(dv);
        sAbsC += fabsf(cv);
        sAbsD += ad;
        sSl1  += (ad < 1.f) ? 0.5f * dv * dv : ad - 0.5f;
        sC2   += cv * cv;
        sXe2  += xv * xv;
    }
    __shared__ float sm[5][256];
    sm[0][tid] = sAbsC; sm[1][tid] = sAbsD; sm[2][tid] = sSl1;
    sm[3][tid] = sC2;   sm[4][tid] = sXe2;
    __syncthreads();
    for (int s = 128; s > 0; s >>= 1) {
        if (tid < s)
            for (int q = 0; q < 5; ++q) sm[q][tid] += sm[q][tid + s];
        __syncthreads();
    }
    if (tid == 0) {
        acc[b]      = sm[0][0];
        acc[64 + b] = sm[1][0];
        atomicAdd(&acc[128], sm[2][0]);
        atomicAdd(&acc[129], sm[3][0]);
        atomicAdd(&acc[130], sm[4][0]);
    }
}

__global__ void zero_acc_kernel(float* acc)
{
    if (threadIdx.x < 576) acc[threadIdx.x] = 0.f;
}

__global__ void finalize_kernel(const float* __restrict__ acc,
                                float* __restrict__ e0, float* __restrict__ e1,
                                float* __restrict__ e2, float* __restrict__ loss)
{
    const int b = threadIdx.x;   // 64 threads
    const int dms[3] = {512, 256, 128};
    float* eo[3] = {e0, e1, e2};
    for (int i = 0; i < 3; ++i) {
        float inv = 1.f / (128.f * (float)dms[i]);
        eo[i][b * 2 + 0] = acc[i * 192 + b] * inv;
        eo[i][b * 2 + 1] = acc[i * 192 + 64 + b] * inv;
    }
    if (b == 0) {
        float L = 0.f;
        for (int i = 0; i < 3; ++i) {
            float cnt = 8192.f * (float)dms[i];
            float sl1 = acc[i * 192 + 128] / cnt;
            float er  = sqrtf(acc[i * 192 + 129]) /
                        (sqrtf(acc[i * 192 + 130]) + 1e-6f) - 1.f;
            L += 0.01f * sl1 + 0.01f * er * er;
        }
        *loss = L;
    }
}

extern "C" void kernel_launch(void* const* d_in, const int* in_sizes, int n_in,
                              void* d_out, int out_size, void* d_ws, size_t ws_size,
                              hipStream_t stream)
{
    (void)in_sizes; (void)n_in; (void)out_size; (void)ws_size;

    const float* x      = (const float*)d_in[0];
    const float* decP[3] = {(const float*)d_in[1], (const float*)d_in[2], (const float*)d_in[3]};
    const float* decU[3] = {(const float*)d_in[4], (const float*)d_in[5], (const float*)d_in[6]};
    const float* puP[3]  = {(const float*)d_in[7], (const float*)d_in[8], (const float*)d_in[9]};
    const float* puU[3]  = {(const float*)d_in[10], (const float*)d_in[11], (const float*)d_in[12]};
    const float* impw    = (const float*)d_in[13];

    float* O = (float*)d_out;
    float* W = (float*)d_ws;

    const long M = 8192;
    const int DM0 = 512, DM1 = 256, DM2 = 128;

    // output layout (flat f32, reference return order)
    float* g0   = O;                 // (8192,512)
    float* g1   = O + 4194304;       // (8192,256)
    float* g2   = O + 6291456;       // (8192,128)
    float* g3   = O + 7340032;       // (8192,128)
    float* rec  = O + 8388608;       // (8192,1024)
    float* cf0  = O + 16777216;      // coeffs d0
    float* cf1  = O + 20971520;      // coeffs d1
    float* cf2  = O + 23068672;      // coeffs d2
    float* cf3  = O + 24117248;      // coeffs c_final
    float* e0   = O + 25165824;
    float* e1   = O + 25165952;
    float* e2   = O + 25166080;
    float* loss = O + 25166208;

    // workspace: c buffers (fwd) reused as rec buffers (inv)
    float* c0buf = W;                // (8192,512) -> recbuf level-1 inverse
    float* c1buf = W + 4194304;      // (8192,256) -> recbuf level-2 inverse
    float* acc   = W + 6291456;      // 3 * 192 floats of accumulators

    zero_acc_kernel<<<1, 576, 0, stream>>>(acc);

    auto lift = [&](const float* X, long xR, long xC, const float* Cf,
                    const float* Ad, long aR, long aC, float sg,
                    float* Ot, long oR, long oC, float* GO, int wi, int DM) {
        dim3 grid(DM / NCOLS, (int)(M / MROWS));
        kan_lift_kernel<<<grid, TPB, 0, stream>>>(X, xR, xC, Cf, Ad, aR, aC, sg,
                                                  Ot, oR, oC, GO, impw, wi, DM);
    };

    // ---------------- forward ----------------
    // level 0: xe = x[::2], xo = x[1::2]
    lift(x,     1024, 2, decP[0], x + 1, 1024, 2, -1.f, cf0, 512, 1, g0, 0, DM0);
    lift(cf0,    512, 1, decU[0], x,     1024, 2, +1.f, c0buf, 512, 1, nullptr, 0, DM0);
    reduce_level_kernel<<<64, 256, 0, stream>>>(cf0, c0buf, x, 1024, 2, DM0, acc + 0);

    // level 1: current = c0buf
    lift(c0buf, 512, 2, decP[1], c0buf + 1, 512, 2, -1.f, cf1, 256, 1, g1, 1, DM1);
    lift(cf1,   256, 1, decU[1], c0buf,     512, 2, +1.f, c1buf, 256, 1, nullptr, 0, DM1);
    reduce_level_kernel<<<64, 256, 0, stream>>>(cf1, c1buf, c0buf, 512, 2, DM1, acc + 192);

    // level 2: current = c1buf ; c_final -> cf3, gated -> g3
    lift(c1buf, 256, 2, decP[2], c1buf + 1, 256, 2, -1.f, cf2, 128, 1, g2, 2, DM2);
    lift(cf2,   128, 1, decU[2], c1buf,     256, 2, +1.f, cf3, 128, 1, g3, 3, DM2);
    reduce_level_kernel<<<64, 256, 0, stream>>>(cf2, cf3, c1buf, 256, 2, DM2, acc + 384);

    finalize_kernel<<<1, 64, 0, stream>>>(acc, e0, e1, e2, loss);

    // ---------------- inverse ----------------
    // level 2: rec_prev = g3 (8192,128) -> c1buf reused as (8192,256) interleaved
    lift(g2,        128, 1, puU[2], g3,  128, 1, -1.f, c1buf,     256, 2, nullptr, 0, DM2); // xe -> even
    lift(c1buf,     256, 2, puP[2], g2,  128, 1, +1.f, c1buf + 1, 256, 2, nullptr, 0, DM2); // xo -> odd

    // level 1: rec_prev = c1buf (8192,256) -> c0buf reused as (8192,512)
    lift(g1,        256, 1, puU[1], c1buf, 256, 1, -1.f, c0buf,     512, 2, nullptr, 0, DM1);
    lift(c0buf,     512, 2, puP[1], g1,    256, 1, +1.f, c0buf + 1, 512, 2, nullptr, 0, DM1);

    // level 0: rec_prev = c0buf (8192,512) -> final rec output (8192,1024)
    lift(g0,        512, 1, puU[0], c0buf, 512, 1, -1.f, rec,     1024, 2, nullptr, 0, DM0);
    lift(rec,      1024, 2, puP[0], g0,    512, 1, +1.f, rec + 1, 1024, 2, nullptr, 0, DM0);
}